// ModelNew_17411797418174
// MI455X (gfx1250) — compile-verified
//
#include <hip/hip_runtime.h>
#include <stdint.h>

// Causal depthwise conv1d (K=4) + conv-state cache update for MI455X (gfx1250).
// Memory-bound (AI ~ 1 FLOP/byte): x tiles staged into LDS with the CDNA5
// Tensor Data Mover (one D# descriptor per block, TENSORcnt-tracked), outputs
// streamed with non-temporal stores.

#define TILE_T 64
#define TILE_D 128   // floats per tile in D
#define HALO   3     // K-1
#define KK     4

typedef float    v4f  __attribute__((ext_vector_type(4)));
typedef uint32_t u32x4 __attribute__((ext_vector_type(4)));
typedef int32_t  i32x8 __attribute__((ext_vector_type(8)));
typedef int32_t  i32x4 __attribute__((ext_vector_type(4)));

#if defined(__has_builtin)
#if __has_builtin(__builtin_amdgcn_tensor_load_to_lds)
#define HAVE_TDM 1
#endif
#endif

__device__ __forceinline__ int clampi(int v, int lo, int hi) {
    return v < lo ? lo : (v > hi ? hi : v);
}

__global__ __launch_bounds__(256) void conv1d_fwd_kernel(
    const float* __restrict__ x,            // [T, D]
    const float* __restrict__ weight,       // [D, 4]
    const float* __restrict__ conv_states,  // [NS, SL, D]
    const int*   __restrict__ qsl,          // [nseq+1]
    const int*   __restrict__ cidx,         // [nseq]
    const int*   __restrict__ ism,          // [nseq]
    const int*   __restrict__ resflag,      // [1]
    float* __restrict__ out,                // [T, D]
    int T, int D, int SL, int nseq)
{
    __shared__ float tile[(TILE_T + HALO) * TILE_D];  // 67*128*4B = 34.3 KB
    __shared__ int s_qsl[32];
    __shared__ int s_ism[16];
    __shared__ int s_cidx[16];

    const int t0  = blockIdx.x * TILE_T;
    const int d0  = blockIdx.y * TILE_D;
    const int tid = threadIdx.x;
    const int D4  = D >> 2;

    if (tid <= nseq) s_qsl[tid] = qsl[tid];
    if (tid >= 32 && tid < 32 + nseq) {
        s_ism[tid - 32]  = ism[tid - 32];
        s_cidx[tid - 32] = cidx[tid - 32];
    }

    // --- Stage x rows [t0-HALO, t0+TILE_T) x cols [d0, d0+TILE_D) into LDS ---
    int gt_start = t0 - HALO;
    int lds_row0 = 0;
    if (gt_start < 0) { lds_row0 = -gt_start; gt_start = 0; }
    int gt_end = t0 + TILE_T; if (gt_end > T) gt_end = T;
    const int nrows = gt_end - gt_start;

#ifdef HAVE_TDM
    // One Tensor Data Mover descriptor per block: 2-D tile, TILE_D 4-byte
    // elements per row, nrows rows, global row stride D. Issued by wave 0.
    if (tid < 32) {
        uint64_t ga = (uint64_t)(uintptr_t)(x + (size_t)gt_start * D + d0);
        // low 32 bits of a generic pointer to __shared__ = LDS byte offset
        uint32_t lds_base = (uint32_t)(uintptr_t)&tile[lds_row0 * TILE_D];

        u32x4 g0;
        g0.x = 1u;                                   // count=1: valid user D#
        g0.y = lds_base;                             // lds_addr (bytes)
        g0.z = (uint32_t)ga;                         // global_addr[31:0]
        g0.w = (uint32_t)(ga >> 32) | (2u << 30);    // global_addr[56:32] | type=2

        i32x8 g1;
        g1[0] = (2 << 16);                           // data_size=2 (4 bytes)
        g1[1] = (TILE_D & 0xFFFF) << 16;             // tensor_dim0[15:0]
        g1[2] = ((TILE_D >> 16) & 0xFFFF)            // tensor_dim0[31:16]
              | ((nrows & 0xFFFF) << 16);            // tensor_dim1[15:0]
        g1[3] = ((nrows >> 16) & 0xFFFF)             // tensor_dim1[31:16]
              | ((TILE_D & 0xFFFF) << 16);           // tile_dim0
        g1[4] = (nrows & 0xFFFF);                    // tile_dim1 (tile_dim2=0)
        g1[5] = D;                                   // tensor_dim0_stride[31:0]
        g1[6] = 0;                                   // stride0[47:32]|stride1[15:0]
        g1[7] = 0;

        i32x4 gz4 = {0, 0, 0, 0};
        i32x8 gz8 = {0, 0, 0, 0, 0, 0, 0, 0};
        // clang-23 / therock form: 6 args (extra zero-filled int32x8 group)
        __builtin_amdgcn_tensor_load_to_lds(g0, g1, gz4, gz4, gz8, 0);
        __builtin_amdgcn_s_wait_tensorcnt(0);
    }
#else
    // Fallback: per-lane gfx1250 async global->LDS b128 copies (ASYNCcnt).
    const int nvec = nrows * (TILE_D / 4);
    for (int base = 0; base < nvec; base += 256) {
        int l = base + tid;
        if (l < nvec) {
            int row = l >> 5;
            int col = l & 31;
            uint32_t ldsOff =
                (uint32_t)(uintptr_t)&tile[(lds_row0 + row) * TILE_D + col * 4];
            uint64_t gaddr =
                (uint64_t)(uintptr_t)(x + (size_t)(gt_start + row) * D + d0 + col * 4);
            asm volatile("global_load_async_to_lds_b128 %0, %1, off"
                         : : "v"(ldsOff), "v"(gaddr) : "memory");
        }
    }
    asm volatile("s_wait_asynccnt 0" ::: "memory");
#endif
    __syncthreads();

    // --- Compute: each thread owns one float4 column group for 8 rows ---
    const int dg = tid & 31;          // float4 column within tile
    const int r0 = (tid >> 5) * 8;    // first of 8 consecutive rows

    const v4f* wt4 = (const v4f*)weight;   // one float4 of taps per channel
    const int cbase = d0 + dg * 4;
    v4f w0 = wt4[cbase + 0];
    v4f w1 = wt4[cbase + 1];
    v4f w2 = wt4[cbase + 2];
    v4f w3 = wt4[cbase + 3];
    // per-tap channel vectors: wt_k[i] = weight[channel i][k]
    v4f wt_0 = {w0.x, w1.x, w2.x, w3.x};   // weight[:,0] (tap j=3)
    v4f wt_1 = {w0.y, w1.y, w2.y, w3.y};   // weight[:,1] (tap j=2)
    v4f wt_2 = {w0.z, w1.z, w2.z, w3.z};   // weight[:,2] (tap j=1)
    v4f wt_3 = {w0.w, w1.w, w2.w, w3.w};   // weight[:,3] (tap j=0)
    const bool res = resflag[0] != 0;

    const v4f* tile4 = (const v4f*)tile;   // row stride = 32 float4
    const v4f* cs4   = (const v4f*)conv_states;
    v4f*       out4  = (v4f*)out;

    v4f xm1 = tile4[(r0 + 2) * 32 + dg];   // x[t-1]
    v4f xm2 = tile4[(r0 + 1) * 32 + dg];   // x[t-2]
    v4f xm3 = tile4[(r0 + 0) * 32 + dg];   // x[t-3]

    // sequence id for first row, then advance incrementally
    int t_first = t0 + r0;
    int s = 0;
    for (int k = 1; k < nseq; ++k)
        if (t_first >= s_qsl[k]) s = k;

    for (int rr = 0; rr < 8; ++rr) {
        int r = r0 + rr;
        int t = t0 + r;
        if (t >= T) break;
        v4f xm0 = tile4[(r + 3) * 32 + dg];   // x[t]

        while (s < nseq - 1 && t >= s_qsl[s + 1]) ++s;
        int pos = t - s_qsl[s];

        v4f s1 = xm1, s2 = xm2, s3 = xm3;
        if (pos < HALO) {  // boundary rows: taps reach into prev state (rare)
            bool hp = s_ism[s] != 0;
            size_t cb = (size_t)s_cidx[s] * SL * D4 + (d0 >> 2) + dg;
            v4f z = {0.f, 0.f, 0.f, 0.f};
            // j-th tap prev index = clip(SL - j + pos, 0, SL-1)
            if (pos < 1) s1 = hp ? cs4[cb + (size_t)clampi(SL - 1 + pos, 0, SL - 1) * D4] : z;
            if (pos < 2) s2 = hp ? cs4[cb + (size_t)clampi(SL - 2 + pos, 0, SL - 1) * D4] : z;
            s3 = hp ? cs4[cb + (size_t)clampi(SL - 3 + pos, 0, SL - 1) * D4] : z;
        }

        v4f o = xm0 * wt_3 + s1 * wt_2 + s2 * wt_1 + s3 * wt_0;
        if (res) o += xm0;

        __builtin_nontemporal_store(o, &out4[(size_t)t * D4 + (d0 >> 2) + dg]);

        xm3 = xm2; xm2 = xm1; xm1 = xm0;
    }
}

__global__ __launch_bounds__(256) void cs_copy_kernel(
    const float* __restrict__ cs, float* __restrict__ outcs, int n4)
{
    int i = blockIdx.x * blockDim.x + threadIdx.x;
    if (i < n4) {
        v4f v = ((const v4f*)cs)[i];
        __builtin_nontemporal_store(v, &((v4f*)outcs)[i]);
    }
}

__global__ __launch_bounds__(256) void cs_update_kernel(
    const float* __restrict__ x, const float* __restrict__ cs,
    const int* __restrict__ qsl, const int* __restrict__ cidx,
    const int* __restrict__ ism, float* __restrict__ outcs,
    int T, int D, int SL, int nseq)
{
    int D4 = D >> 2;
    int total = nseq * SL * D4;
    int i = blockIdx.x * blockDim.x + threadIdx.x;
    if (i >= total) return;
    int dg  = i % D4;
    int rem = i / D4;
    int si  = rem % SL;
    int s   = rem / SL;

    int start = qsl[s];
    int L = qsl[s + 1] - start;
    int p = L - SL + si;

    v4f v;
    if (p >= 0) {
        int g = clampi(start + p, 0, T - 1);
        v = ((const v4f*)x)[(size_t)g * D4 + dg];
    } else if (ism[s] != 0) {
        int pi = clampi(p + SL, 0, SL - 1);
        v = ((const v4f*)cs)[((size_t)cidx[s] * SL + pi) * D4 + dg];
    } else {
        v = (v4f){0.f, 0.f, 0.f, 0.f};
    }
    ((v4f*)outcs)[((size_t)cidx[s] * SL + si) * D4 + dg] = v;
}

extern "C" void kernel_launch(void* const* d_in, const int* in_sizes, int n_in,
                              void* d_out, int out_size, void* d_ws, size_t ws_size,
                              hipStream_t stream) {
    (void)n_in; (void)out_size; (void)d_ws; (void)ws_size;
    const float* x    = (const float*)d_in[0];
    const float* w    = (const float*)d_in[1];
    const float* cs   = (const float*)d_in[2];
    const int*   qsl  = (const int*)d_in[3];
    const int*   cidx = (const int*)d_in[4];
    const int*   ism  = (const int*)d_in[5];
    const int*   res  = (const int*)d_in[7];   // residual_connection flag

    const int DK   = in_sizes[1];          // D * K
    const int D    = DK / KK;              // 2048
    const int T    = in_sizes[0] / D;      // 16384
    const int nseq = in_sizes[3] - 1;      // 8
    const int SL   = KK - 1;               // 3
    const int NS   = in_sizes[2] / (SL * D);

    float* out   = (float*)d_out;
    float* outcs = out + (size_t)T * D;

    // Main conv: 2-D grid (no runtime division): x = T tiles, y = D tiles.
    dim3 grid((T + TILE_T - 1) / TILE_T, D / TILE_D);
    conv1d_fwd_kernel<<<grid, 256, 0, stream>>>(
        x, w, cs, qsl, cidx, ism, res, out, T, D, SL, nseq);

    // Copy all conv_states to output region, then overwrite cached slots.
    const int n4 = NS * SL * (D / 4);
    cs_copy_kernel<<<(n4 + 255) / 256, 256, 0, stream>>>(cs, outcs, n4);

    const int nu = nseq * SL * (D / 4);
    cs_update_kernel<<<(nu + 255) / 256, 256, 0, stream>>>(
        x, cs, qsl, cidx, ism, outcs, T, D, SL, nseq);
}